// GraphPool_29746943492200
// MI455X (gfx1250) — compile-verified
//
#include <hip/hip_runtime.h>

// ---------------------------------------------------------------------------
// Graph max-pool, degree-binned.  N=1M atoms, D=128 f32 feats (512 B/row).
// Pure bandwidth problem (~3.7 GB @ 23.3 TB/s ~= 160 us). One wave32 per
// gathered row: lane l owns bytes [16l, 16l+16) of the 512 B row (b128/lane).
// Neighbor rows staged via CDNA5 async-to-LDS (ASYNCcnt) engine.
//
// L2 policy (192 MB): random gathers over the 512 MB feature table are the
// only re-referenced stream -> keep them (and self-row reads, which feed the
// same table) regular-temporal; make write-once output stores and read-once
// adjacency loads non-temporal so they don't evict gather-resident lines.
// ---------------------------------------------------------------------------

#define MAX_DEG   10
#define D4        32          // 128 floats = 32 float4 per row
#define ROW_BYTES 512
#define WAVES_PER_BLOCK 8     // 256 threads = 8 waves (wave32)

typedef float v4f __attribute__((ext_vector_type(4)));

struct PoolParams {
    const int* adj[MAX_DEG];  // adj[d-1]: [count_d, d] int32 neighbor indices
    int start[MAX_DEG];       // start[d-1]: first global row of degree-d bin
};

__device__ __forceinline__ v4f max4(v4f a, v4f b) {
    v4f r;
    r.x = fmaxf(a.x, b.x);
    r.y = fmaxf(a.y, b.y);
    r.z = fmaxf(a.z, b.z);
    r.w = fmaxf(a.w, b.w);
    return r;
}

// CDNA5 async global->LDS copy, 16 B per active lane. Tracked by ASYNCcnt.
// LDS destination VGPR holds the 32-bit LDS byte address (low 32 bits of the
// flat shared-aperture pointer, per ISA 10.2 aperture mapping).
__device__ __forceinline__ void async_ld_b128(const void* gaddr, unsigned lds_byte) {
    asm volatile("global_load_async_to_lds_b128 %0, %1, off"
                 :: "v"(lds_byte), "v"((unsigned long long)gaddr)
                 : "memory");
}
__device__ __forceinline__ void wait_async0() {
    asm volatile("s_wait_asynccnt 0" ::: "memory");
}

// ---- Kernel A: degree-0 rows pass through (streaming copy) ----------------
// Input rows stay RT (they are gather targets for kernel B); output is NT.
__global__ void pool_passthrough(const v4f* __restrict__ in,
                                 v4f* __restrict__ out, long n4) {
    long stride = (long)gridDim.x * blockDim.x;
    for (long i = (long)blockIdx.x * blockDim.x + threadIdx.x; i < n4; i += stride) {
        __builtin_prefetch(in + i + stride, 0, 3);   // -> global_prefetch_b8
        v4f v = in[i];
        __builtin_nontemporal_store(v, out + i);     // th:TH_STORE_NT
    }
}

// ---- Kernel B: one wave per degree>=1 row ---------------------------------
__global__ void pool_gather_max(const v4f* __restrict__ feats,
                                v4f* __restrict__ out,
                                PoolParams p, int count0, int n_rows) {
    // [wave][neighbor k][lane] staging, 512 B per neighbor row, 40 KB/block.
    __shared__ v4f stage[WAVES_PER_BLOCK][MAX_DEG * 32];

    const int lane  = threadIdx.x & 31;
    const int wslot = threadIdx.x >> 5;
    const int w     = blockIdx.x * WAVES_PER_BLOCK + wslot;   // wave-uniform
    const int r     = count0 + w;                             // global row
    if (r >= n_rows) return;                                  // wave-uniform

    // Locate degree bin (uniform kernel-arg starts; 9 compares).
    int d = 1;
#pragma unroll
    for (int t = 2; t <= MAX_DEG; ++t)
        d = (r >= p.start[t - 1]) ? t : d;
    const int j = r - p.start[d - 1];
    const int* __restrict__ adj = p.adj[d - 1];

    // Lanes 0..d-1 fetch the neighbor indices (read-once -> NT); shfl-bcast.
    int my_idx = 0;
    if (lane < d)
        my_idx = __builtin_nontemporal_load(adj + j * d + lane);

    // Fire async stages: one b128/lane per neighbor row (512 B/row, RT so
    // hot rows stay L2-resident for other waves' gathers).
    const char* fb = (const char*)feats;
    unsigned lds_lane =
        (unsigned)(unsigned long long)(const void*)&stage[wslot][0] + lane * 16u;
    for (int k = 0; k < d; ++k) {
        int idx = __shfl(my_idx, k);
        async_ld_b128(fb + (size_t)idx * ROW_BYTES + lane * 16,
                      lds_lane + (unsigned)(k * ROW_BYTES));
    }

    // Self row loads directly into VGPRs, overlapping the async transfers.
    v4f acc = feats[(size_t)r * D4 + lane];

    wait_async0();   // s_wait_asynccnt 0

    for (int k = 0; k < d; ++k)
        acc = max4(acc, stage[wslot][k * 32 + lane]);

    __builtin_nontemporal_store(acc, out + (size_t)r * D4 + lane);  // NT
}

// ---------------------------------------------------------------------------
extern "C" void kernel_launch(void* const* d_in, const int* in_sizes, int n_in,
                              void* d_out, int out_size, void* d_ws, size_t ws_size,
                              hipStream_t stream) {
    (void)n_in; (void)out_size; (void)d_ws; (void)ws_size;

    const float* feats = (const float*)d_in[0];
    float* out = (float*)d_out;
    const int N = in_sizes[0] / 128;

    // d_in[1] is deg_slice (unused: bin layout is derivable from adj sizes).
    PoolParams p;
    long total_binned = 0;
    int counts[MAX_DEG + 1];
    for (int d = 1; d <= MAX_DEG; ++d) {
        counts[d] = in_sizes[1 + d] / d;        // adj_d has shape [count_d, d]
        p.adj[d - 1] = (const int*)d_in[1 + d];
        total_binned += counts[d];
    }
    const int count0 = (int)(N - total_binned);
    int s = count0;
    for (int d = 1; d <= MAX_DEG; ++d) { p.start[d - 1] = s; s += counts[d]; }

    // Kernel A: copy degree-0 prefix (count0 rows = count0*32 v4f).
    const long n4 = (long)count0 * D4;
    if (n4 > 0)
        pool_passthrough<<<1024, 256, 0, stream>>>(
            (const v4f*)feats, (v4f*)out, n4);

    // Kernel B: one wave32 per remaining row.
    const int rows = N - count0;
    if (rows > 0) {
        const int blocks = (rows + WAVES_PER_BLOCK - 1) / WAVES_PER_BLOCK;
        pool_gather_max<<<blocks, WAVES_PER_BLOCK * 32, 0, stream>>>(
            (const v4f*)feats, (v4f*)out, p, count0, N);
    }
}